// AttentionRouter_72602127171974
// MI455X (gfx1250) — compile-verified
//
#include <hip/hip_runtime.h>

typedef __attribute__((ext_vector_type(2))) float v2f;
typedef __attribute__((ext_vector_type(4))) float v4f;
typedef __attribute__((ext_vector_type(8))) float v8f;

#define TT 32768
#define BB 8
#define HH 8
#define DD 128
#define HD 1024 /* H*D */
#define ROWS_PER_BLK 64

// ---------------- kernel 0: zero the segment-sum workspace ----------------
__global__ void zero_ws_kernel(float* __restrict__ ws, int n) {
  int i = blockIdx.x * blockDim.x + threadIdx.x;
  if (i < n) ws[i] = 0.0f;
}

// ---------------- kernel 1: segment sum over T ----------------------------
__device__ __forceinline__ int seg_of(int t, int c1, int c2, int c3, int c4,
                                      int c5, int c6, int c7) {
  return (c1 <= t) + (c2 <= t) + (c3 <= t) + (c4 <= t) +
         (c5 <= t) + (c6 <= t) + (c7 <= t);
}

// grid 512 x 256 threads; each block handles 64 consecutive rows of (T,1024)
__global__ void seg_sum_kernel(const float* __restrict__ x,
                               const int* __restrict__ cu,
                               float* __restrict__ sums) {
  const int tid  = threadIdx.x;      // 0..255
  const int col  = tid << 2;         // 4 floats per thread
  const int row0 = blockIdx.x * ROWS_PER_BLK;

  const int c1 = cu[1], c2 = cu[2], c3 = cu[3], c4 = cu[4],
            c5 = cu[5], c6 = cu[6], c7 = cu[7];

  const int seg_first = seg_of(row0, c1, c2, c3, c4, c5, c6, c7);
  const int seg_last  = seg_of(row0 + ROWS_PER_BLK - 1, c1, c2, c3, c4, c5, c6, c7);

  if (seg_first == seg_last) {
    // Fast path (>= 505 of 512 blocks): branch-free streaming accumulate,
    // 4 independent accumulators so loads pipeline freely.
    const float* p = x + (size_t)row0 * HD + col;
    v4f a0 = {}, a1 = {}, a2 = {}, a3 = {};
#pragma unroll
    for (int r = 0; r < ROWS_PER_BLK; r += 4) {
      a0 += *(const v4f*)(p + (size_t)(r + 0) * HD);
      a1 += *(const v4f*)(p + (size_t)(r + 1) * HD);
      a2 += *(const v4f*)(p + (size_t)(r + 2) * HD);
      a3 += *(const v4f*)(p + (size_t)(r + 3) * HD);
    }
    const v4f acc = (a0 + a1) + (a2 + a3);
    float* dst = sums + seg_first * HD + col;
    atomicAdd(dst + 0, acc.x); atomicAdd(dst + 1, acc.y);
    atomicAdd(dst + 2, acc.z); atomicAdd(dst + 3, acc.w);
    return;
  }

  // Slow path: block straddles a segment boundary (rare).
  v4f acc = {};
  int cur = -1;
  for (int r = 0; r < ROWS_PER_BLK; ++r) {
    const int t = row0 + r;
    const int seg = seg_of(t, c1, c2, c3, c4, c5, c6, c7);
    if (seg != cur) {               // uniform across block (same t range)
      if (cur >= 0) {
        float* dst = sums + cur * HD + col;
        atomicAdd(dst + 0, acc.x); atomicAdd(dst + 1, acc.y);
        atomicAdd(dst + 2, acc.z); atomicAdd(dst + 3, acc.w);
      }
      cur = seg;
      acc = (v4f){0.f, 0.f, 0.f, 0.f};
    }
    acc += *(const v4f*)(x + (size_t)t * HD + col);
  }
  float* dst = sums + cur * HD + col;
  atomicAdd(dst + 0, acc.x); atomicAdd(dst + 1, acc.y);
  atomicAdd(dst + 2, acc.z); atomicAdd(dst + 3, acc.w);
}

// ---------------- kernel 2: pooled mean + WMMA MLP + argmax ---------------
__device__ __forceinline__ v8f wmma_f32_k4(v2f a, v2f b, v8f c) {
  // 8 args: (neg_a, A, neg_b, B, c_mod, C, reuse_a, reuse_b)
  return __builtin_amdgcn_wmma_f32_16x16x4_f32(
      false, a, false, b, (short)0, c, false, false);
}

// One 16xN layer:  O = act(A[16xK] @ W[KxN] + bias), A/O in LDS, W in global.
template <bool SILU>
__device__ __forceinline__ void mlp_layer(const float* __restrict__ A,
                                          const float* __restrict__ W,
                                          const float* __restrict__ bias,
                                          float* __restrict__ O,
                                          int K, int N,
                                          int wave, int nwaves, int lane) {
  const int m    = lane & 15;
  const int koff = (lane >> 4) << 1;   // 0 or 2  (A/B K-half per lane group)
  const int nl   = lane & 15;
  const int mh   = (lane >> 4) << 3;   // 0 or 8  (C/D row-half per lane group)
  for (int tile = wave; tile < (N >> 4); tile += nwaves) {
    const int n0 = tile << 4;
    const float* Wp = W + n0 + nl;
    v8f acc = {};
    for (int k0 = 0; k0 < K; k0 += 4) {
      v2f a, b;
      a.x = A[m * K + k0 + koff];
      a.y = A[m * K + k0 + koff + 1];
      b.x = Wp[(size_t)(k0 + koff) * N];
      b.y = Wp[(size_t)(k0 + koff + 1) * N];
      acc = wmma_f32_k4(a, b, acc);
    }
    const float bn = bias[n0 + nl];
#pragma unroll
    for (int r = 0; r < 8; ++r) {
      float v = acc[r] + bn;
      if (SILU) v = v / (1.0f + __expf(-v));
      O[(r + mh) * N + n0 + nl] = v;
    }
  }
}

__global__ void mlp_kernel(const float* __restrict__ sums,
                           const int* __restrict__ cu,
                           const float* __restrict__ w1, const float* __restrict__ b1,
                           const float* __restrict__ w2, const float* __restrict__ b2,
                           const float* __restrict__ w3, const float* __restrict__ b3,
                           const float* __restrict__ w4, const float* __restrict__ b4,
                           const float* __restrict__ w5, const float* __restrict__ b5,
                           float* __restrict__ out) {
  extern __shared__ float smem[];
  float* act0 = smem;               // [16][128]
  float* act1 = act0 + 16 * 128;    // [16][1024]
  float* act2 = act1 + 16 * 1024;   // [16][256]
  float* act3 = act2 + 16 * 256;    // [16][512]
  float* act4 = act3 + 16 * 512;    // [16][128]

  const int tid   = threadIdx.x;    // 1024 threads = 32 wave32
  const int lane  = tid & 31;
  const int wave  = tid >> 5;
  const int nwav  = blockDim.x >> 5;

  // pooled_mean into act0, rows 8..15 zero-padded
  for (int i = tid; i < 16 * DD; i += blockDim.x) {
    const int b = i >> 7, d = i & 127;
    float v = 0.0f;
    if (b < BB) {
      float s = 0.0f;
#pragma unroll
      for (int h = 0; h < HH; ++h) s += sums[b * HD + h * DD + d];
      const int cnt = cu[b + 1] - cu[b];
      const float c = (cnt < 1) ? 1.0f : (float)cnt;
      v = s / (c * (float)HH);
    }
    act0[i] = v;
  }
  __syncthreads();

  mlp_layer<true >(act0, w1, b1, act1, 128, 1024, wave, nwav, lane);  // silu
  __syncthreads();
  mlp_layer<false>(act1, w2, b2, act2, 1024, 256, wave, nwav, lane);  // linear
  __syncthreads();
  mlp_layer<true >(act2, w3, b3, act3, 256, 512, wave, nwav, lane);   // silu
  __syncthreads();
  mlp_layer<true >(act3, w4, b4, act4, 512, 128, wave, nwav, lane);   // silu
  __syncthreads();

  // final 128 -> 2 logits, argmax (softmax is monotone; ties -> index 0)
  if (tid < BB) {
    const float* a = act4 + tid * DD;
    float l0 = b5[0], l1 = b5[1];
    for (int k = 0; k < DD; ++k) {
      const float av = a[k];
      l0 += av * w5[2 * k + 0];
      l1 += av * w5[2 * k + 1];
    }
    const float z = (l1 > l0) ? 1.0f : 0.0f;
#pragma unroll
    for (int h = 0; h < HH; ++h) out[tid * HH + h] = z;
  }
}

extern "C" void kernel_launch(void* const* d_in, const int* in_sizes, int n_in,
                              void* d_out, int out_size, void* d_ws, size_t ws_size,
                              hipStream_t stream) {
  const float* x  = (const float*)d_in[0];
  const int*   cu = (const int*)  d_in[1];
  const float* w1 = (const float*)d_in[2];
  const float* b1 = (const float*)d_in[3];
  const float* w2 = (const float*)d_in[4];
  const float* b2 = (const float*)d_in[5];
  const float* w3 = (const float*)d_in[6];
  const float* b3 = (const float*)d_in[7];
  const float* w4 = (const float*)d_in[8];
  const float* b4 = (const float*)d_in[9];
  const float* w5 = (const float*)d_in[10];
  const float* b5 = (const float*)d_in[11];
  float* sums = (float*)d_ws;               // 8 * 1024 floats
  float* out  = (float*)d_out;              // 64 floats

  zero_ws_kernel<<<(BB * HD + 255) / 256, 256, 0, stream>>>(sums, BB * HD);
  seg_sum_kernel<<<TT / ROWS_PER_BLK, 256, 0, stream>>>(x, cu, sums);
  const size_t lds_bytes = (size_t)(16 * 128 + 16 * 1024 + 16 * 256 +
                                    16 * 512 + 16 * 128) * sizeof(float); // 128 KB
  mlp_kernel<<<1, 1024, lds_bytes, stream>>>(sums, cu, w1, b1, w2, b2, w3, b3,
                                             w4, b4, w5, b5, out);
}